// RotatedIoULoss_85048942395432
// MI455X (gfx1250) — compile-verified
//
#include <hip/hip_runtime.h>
#include <math.h>

// Rotated-IoU loss for MI455X (gfx1250, wave32).
// VALU/transcendental-bound elementwise kernel + deterministic 2-pass mean.
// Reductions use V_WMMA_F32_16X16X4_F32 (B = ones) on the matrix pipe.

#define NVERT 24
#define NSORT 32
#define BLOCK 256

typedef __attribute__((ext_vector_type(2))) float v2f;
typedef __attribute__((ext_vector_type(8))) float v8f;

__device__ __forceinline__ float xprod(float ax, float ay, float bx, float by) {
    return ax * by - ay * bx;
}

// Monotone surrogate of atan2(y,x) on (-pi,pi] -> [-2,2]. Preserves ordering,
// replaces 24 atan2f calls per box with ~6 VALU ops each.
__device__ __forceinline__ float pseudo_angle(float x, float y) {
    float r = y / fmaxf(fabsf(x) + fabsf(y), 1e-30f);
    return (x >= 0.0f) ? r : ((y >= 0.0f) ? (2.0f - r) : (-2.0f - r));
}

// Sum of the 64 values {x0,x1} across all 32 lanes of the wave, returned to
// every lane. Uses D = A(16x4) * ones(4x16): each D row holds a row-sum of A;
// summing the 8 C/D VGPRs per lane covers 8 rows, the xor-16 shuffle adds the
// other 8. Total equals sum of all A entries independent of the A/B lane
// mapping, so this is layout-proof. EXEC must be all 1s at the call site.
__device__ __forceinline__ float wave_sum_wmma(float x0, float x1) {
    v2f a; a.x = x0; a.y = x1;
    v2f b; b.x = 1.0f; b.y = 1.0f;
    v8f c = {};
    c = __builtin_amdgcn_wmma_f32_16x16x4_f32(
        /*neg_a=*/false, a, /*neg_b=*/false, b,
        /*c_mod=*/(short)0, c, /*reuse_a=*/false, /*reuse_b=*/false);
    float s = ((c[0] + c[1]) + (c[2] + c[3])) + ((c[4] + c[5]) + (c[6] + c[7]));
    s += __shfl_xor(s, 16, 32);
    return s;
}

__global__ __launch_bounds__(BLOCK)
void riou_partial(const float* __restrict__ pred,
                  const float* __restrict__ target,
                  float* __restrict__ partials, int n) {
    const int idx = blockIdx.x * BLOCK + threadIdx.x;
    const int cid = (idx < n) ? idx : (n - 1);   // clamp; result masked below
    const float act = (idx < n) ? 1.0f : 0.0f;   // no divergence: EXEC stays full

    const float* bp = pred   + 5 * (long long)cid;
    const float* bt = target + 5 * (long long)cid;
    // Streamed once -> non-temporal loads (TH=NT); compiler merges to b128+b32.
    const float pcx = __builtin_nontemporal_load(bp + 0);
    const float pcy = __builtin_nontemporal_load(bp + 1);
    const float pw  = __builtin_nontemporal_load(bp + 2);
    const float ph  = __builtin_nontemporal_load(bp + 3);
    const float pa  = __builtin_nontemporal_load(bp + 4);
    const float tcx = __builtin_nontemporal_load(bt + 0);
    const float tcy = __builtin_nontemporal_load(bt + 1);
    const float tw  = __builtin_nontemporal_load(bt + 2);
    const float th  = __builtin_nontemporal_load(bt + 3);
    const float ta  = __builtin_nontemporal_load(bt + 4);

    float sp, cp, st, ct;
    __sincosf(pa, &sp, &cp);
    __sincosf(ta, &st, &ct);

    // --- corners of both boxes (template * wh, rotate, translate) ---
    const float tplx[4] = { 0.5f, -0.5f, -0.5f,  0.5f };
    const float tply[4] = { 0.5f,  0.5f, -0.5f, -0.5f };
    float c1x[4], c1y[4], c2x[4], c2y[4];
#pragma unroll
    for (int i = 0; i < 4; ++i) {
        float px = tplx[i] * pw, py = tply[i] * ph;
        c1x[i] = px * cp - py * sp + pcx;
        c1y[i] = px * sp + py * cp + pcy;
        float qx = tplx[i] * tw, qy = tply[i] * th;
        c2x[i] = qx * ct - qy * st + tcx;
        c2y[i] = qx * st + qy * ct + tcy;
    }

    // --- candidate vertices: 8 corners + 16 edge intersections ---
    bool  m[NVERT];
    float vx[NVERT], vy[NVERT];
    const float tol = 1e-6f;
#pragma unroll
    for (int i = 0; i < 4; ++i) {              // c1 corners inside target
        float px = c1x[i] - tcx, py = c1y[i] - tcy;
        float lx =  px * ct + py * st;
        float ly = -px * st + py * ct;
        m[i]  = (fabsf(lx) <= tw * 0.5f + tol) && (fabsf(ly) <= th * 0.5f + tol);
        vx[i] = c1x[i]; vy[i] = c1y[i];
    }
#pragma unroll
    for (int i = 0; i < 4; ++i) {              // c2 corners inside pred
        float px = c2x[i] - pcx, py = c2y[i] - pcy;
        float lx =  px * cp + py * sp;
        float ly = -px * sp + py * cp;
        m[4 + i]  = (fabsf(lx) <= pw * 0.5f + tol) && (fabsf(ly) <= ph * 0.5f + tol);
        vx[4 + i] = c2x[i]; vy[4 + i] = c2y[i];
    }
#pragma unroll
    for (int i = 0; i < 4; ++i) {
        const int ii = (i + 1) & 3;
        const float rx = c1x[ii] - c1x[i], ry = c1y[ii] - c1y[i];
#pragma unroll
        for (int j = 0; j < 4; ++j) {
            const int jj = (j + 1) & 3;
            const float sx = c2x[jj] - c2x[j], sy = c2y[jj] - c2y[j];
            const float den = xprod(rx, ry, sx, sy);
            const float ad = fabsf(den);
            const float safe = (ad < 1e-12f) ? 1e-12f : den;
            const float inv = 1.0f / safe;           // one rcp, reused for t & u
            const float qpx = c2x[j] - c1x[i], qpy = c2y[j] - c1y[i];
            const float t = xprod(qpx, qpy, sx, sy) * inv;
            const float u = xprod(qpx, qpy, rx, ry) * inv;
            const bool ok = (ad > 1e-12f) && (t >= 0.0f) && (t <= 1.0f)
                                          && (u >= 0.0f) && (u <= 1.0f);
            const int v = 8 + i * 4 + j;
            vx[v] = c1x[i] + t * rx;
            vy[v] = c1y[i] + t * ry;
            m[v]  = ok;
        }
    }

    // --- masked centroid ---
    float sxv = 0.0f, syv = 0.0f; int nv = 0;
#pragma unroll
    for (int i = 0; i < NVERT; ++i) {
        float mf = m[i] ? 1.0f : 0.0f;
        sxv += vx[i] * mf; syv += vy[i] * mf; nv += m[i] ? 1 : 0;
    }
    const float invnv = 1.0f / fmaxf((float)nv, 1.0f);
    const float ctrx = sxv * invnv, ctry = syv * invnv;

    // --- sort keys (pseudo-angle), payload = centered masked vertex ---
    float key[NSORT], dx[NSORT], dy[NSORT];
#pragma unroll
    for (int i = 0; i < NVERT; ++i) {
        float mf = m[i] ? 1.0f : 0.0f;
        float ddx = (vx[i] - ctrx) * mf;
        float ddy = (vy[i] - ctry) * mf;
        dx[i] = ddx; dy[i] = ddy;
        key[i] = m[i] ? pseudo_angle(ddx, ddy) : 1e8f;   // invalid -> big
    }
#pragma unroll
    for (int i = NVERT; i < NSORT; ++i) { key[i] = 1e9f; dx[i] = 0.0f; dy[i] = 0.0f; }

    // --- register-resident bitonic sort, fully unrolled (no scratch) ---
#pragma unroll
    for (int k = 2; k <= NSORT; k <<= 1) {
#pragma unroll
        for (int j = k >> 1; j > 0; j >>= 1) {
#pragma unroll
            for (int i = 0; i < NSORT; ++i) {
                const int l = i ^ j;
                if (l > i && i < NVERT) {   // comparators among padding pruned
                    const bool up = ((i & k) == 0);
                    float ka = key[i], kb = key[l];
                    bool sw = up ? (ka > kb) : (ka < kb);
                    key[i] = sw ? kb : ka;  key[l] = sw ? ka : kb;
                    float xa = dx[i], xb = dx[l];
                    dx[i] = sw ? xb : xa;   dx[l] = sw ? xa : xb;
                    float ya = dy[i], yb = dy[l];
                    dy[i] = sw ? yb : ya;   dy[l] = sw ? ya : yb;
                }
            }
        }
    }

    // --- shoelace over the 24 cyclic pairs (invalid slots have d = 0) ---
    float part = 0.0f;
#pragma unroll
    for (int i = 0; i < NVERT; ++i) {
        const int j = (i + 1) % NVERT;
        part += xprod(dx[i], dy[i], dx[j], dy[j]);
    }
    // closing term: last VALID vertex (predicated scan, no dynamic indexing)
    float lvx = dx[0], lvy = dy[0];
#pragma unroll
    for (int i = 0; i < NVERT; ++i) {
        bool valid = key[i] < 1e7f;
        lvx = valid ? dx[i] : lvx;
        lvy = valid ? dy[i] : lvy;
    }
    float inter = 0.5f * fabsf(part + xprod(lvx, lvy, dx[0], dy[0]));
    inter = (nv >= 3) ? inter : 0.0f;

    const float a1 = pw * ph, a2 = tw * th;
    const float uni = fmaxf(a1 + a2 - inter, 1e-12f);
    const float iou = fmaxf(inter / uni, 1e-6f);
    const float loss = -__logf(iou) * act;

    // --- per-wave reduction on the matrix pipe + LDS across the 8 waves ---
    const float wsum = wave_sum_wmma(loss, 0.0f);
    __shared__ float red[BLOCK / 32];
    const int lane = threadIdx.x & 31;
    const int wid  = threadIdx.x >> 5;
    if (lane == 0) red[wid] = wsum;
    __syncthreads();
    if (threadIdx.x == 0) {
        float s = 0.0f;
#pragma unroll
        for (int i = 0; i < BLOCK / 32; ++i) s += red[i];
        partials[blockIdx.x] = s;
    }
}

__global__ __launch_bounds__(BLOCK)
void riou_finalize(const float* __restrict__ partials, int nparts,
                   float* __restrict__ out, float invN) {
    // Uniform trip count + predicated loads: EXEC stays all-1s for the WMMA.
    float s = 0.0f;
    const int iters = (nparts + BLOCK - 1) / BLOCK;
    for (int it = 0; it < iters; ++it) {
        const int i = it * BLOCK + threadIdx.x;
        s += (i < nparts) ? partials[i] : 0.0f;
    }
    const float wsum = wave_sum_wmma(s, 0.0f);
    __shared__ float red[BLOCK / 32];
    const int lane = threadIdx.x & 31;
    const int wid  = threadIdx.x >> 5;
    if (lane == 0) red[wid] = wsum;
    __syncthreads();
    if (threadIdx.x == 0) {
        float t = 0.0f;
#pragma unroll
        for (int i = 0; i < BLOCK / 32; ++i) t += red[i];
        out[0] = t * invN;   // LOSS_WEIGHT == 1.0
    }
}

extern "C" void kernel_launch(void* const* d_in, const int* in_sizes, int n_in,
                              void* d_out, int out_size, void* d_ws, size_t ws_size,
                              hipStream_t stream) {
    const float* pred   = (const float*)d_in[0];
    const float* target = (const float*)d_in[1];
    float* out = (float*)d_out;
    float* partials = (float*)d_ws;

    const int n = in_sizes[0] / 5;                 // (N,5) float32
    const int nblocks = (n + BLOCK - 1) / BLOCK;   // 1954 for N=500000 (~7.8 KB ws)

    riou_partial<<<nblocks, BLOCK, 0, stream>>>(pred, target, partials, n);
    riou_finalize<<<1, BLOCK, 0, stream>>>(partials, nblocks, out, 1.0f / (float)n);
}